// Zim_67430986547716
// MI455X (gfx1250) — compile-verified
//
#include <hip/hip_runtime.h>

typedef float v2f __attribute__((ext_vector_type(2)));
typedef float v8f __attribute__((ext_vector_type(8)));

#define MASKN 64
#define NPOINT 32
#define PADW 128                          // 64 real entries + 64 zeros (for lanes 16-31)
#define INV_TWO_SIGMA2 (1.0f / 882.0f)    // 2*21*21
#define INV_STRIDE (1.0f / 16.0f)

__global__ __launch_bounds__(128) void Zim_sam_mask_kernel(
    const float* __restrict__ boxes,      // (B,4)
    const float* __restrict__ pts,        // (B,32,2)
    float* __restrict__ out,              // bbox (B,64,64) then point (B,64,64)
    int B)
{
    __shared__ float s_ex[NPOINT * PADW]; // e_x[p][j], j>=64 -> 0
    __shared__ float s_ey[NPOINT * PADW]; // e_y[p][i], i>=64 -> 0
    __shared__ float s_inx[PADW];         // upper half zero
    __shared__ float s_iny[PADW];         // upper half zero
    __shared__ int   s_px[NPOINT];
    __shared__ int   s_py[NPOINT];
    __shared__ float s_validf[NPOINT];

    const int b    = blockIdx.x;
    const int t    = threadIdx.x;
    const int lane = t & 31;
    const int wave = t >> 5;

    // ---- Stage 0: quantize points (threads 0..31) + box indicators ----
    if (t < NPOINT) {
        float fx = pts[(size_t)(b * NPOINT + t) * 2 + 0];
        float fy = pts[(size_t)(b * NPOINT + t) * 2 + 1];
        int px = (int)floorf(fx * INV_STRIDE);
        int py = (int)floorf(fy * INV_STRIDE);
        bool valid = (px >= 0) && (py >= 0) && (px < MASKN) && (py < MASKN);
        px = px > (1 << 20) ? (1 << 20) : (px < -(1 << 20) ? -(1 << 20) : px);
        py = py > (1 << 20) ? (1 << 20) : (py < -(1 << 20) ? -(1 << 20) : py);
        s_px[t] = px;
        s_py[t] = py;
        s_validf[t] = valid ? 1.0f : 0.0f;
    }
    {
        float x0 = boxes[(size_t)b * 4 + 0];
        float y0 = boxes[(size_t)b * 4 + 1];
        float x1 = boxes[(size_t)b * 4 + 2];
        float y1 = boxes[(size_t)b * 4 + 3];
        int j = t & 63;
        float lo = (t < 64) ? fminf(x0, x1) : fminf(y0, y1);
        float hi = (t < 64) ? fmaxf(x0, x1) : fmaxf(y0, y1);
        int lo_i = (int)floorf(lo * INV_STRIDE);
        int hi_i = (int)floorf(hi * INV_STRIDE) + 1;
        lo_i = lo_i > 0 ? lo_i : 0;
        hi_i = hi_i < MASKN ? hi_i : MASKN;
        float v = (j >= lo_i && j < hi_i) ? 1.0f : 0.0f;
        if (t < 64) { s_inx[j] = v; s_inx[64 + j] = 0.0f; }
        else        { s_iny[j] = v; s_iny[64 + j] = 0.0f; }
    }
    __syncthreads();

    // ---- Stage 1: separable Gaussian factor tables (zero-padded halves) ----
    // idx mod 128 == t, so each thread's j is fixed; p sweeps 0..31.
    for (int idx = t; idx < NPOINT * PADW; idx += 128) {
        int p = idx >> 7;
        int j = idx & 127;
        float ex, ey;
        if (j < 64) {
            float dx = (float)(j - s_px[p]);
            float dy = (float)(j - s_py[p]);
            // fold validity into the x-factor: product is exactly 0 for invalid points
            ex = s_validf[p] * __expf(-(dx * dx) * INV_TWO_SIGMA2);
            ey =               __expf(-(dy * dy) * INV_TWO_SIGMA2);
        } else {
            ex = 0.0f;
            ey = 0.0f;
        }
        s_ex[idx] = ex;
        s_ey[idx] = ey;
    }
    __syncthreads();

    // ---- Stage 2: per-wave tile row; rank-1 outer products via WMMA ----
    const int ti   = wave;           // tile row 0..3
    const int half = lane >> 4;      // 0: D rows M=0..7, 1: M=8..15
    const int lj   = lane & 15;
    const int offA = half * 64 + ti * 16 + lj;   // lanes 16-31 land in zero region

    v2f a_box;
    a_box.x = s_iny[offA];           // unconditional LDS load, no exec masking
    a_box.y = 0.0f;

    const v8f czero = {};
    float* ob = out + (size_t)b * (MASKN * MASKN);
    float* op = out + (size_t)B * (MASKN * MASKN) + (size_t)b * (MASKN * MASKN);

    for (int tj = 0; tj < 4; ++tj) {
        const int offB = half * 64 + tj * 16 + lj;

        v2f b_box;
        b_box.x = s_inx[offB];
        b_box.y = 0.0f;

        // bbox tile = in_y ⊗ in_x
        v8f dbox = __builtin_amdgcn_wmma_f32_16x16x4_f32(
            false, a_box, false, b_box, (short)0, czero, false, false);

        // point tile = max_p (e_y[p] ⊗ e_x[p])
        v8f m = {};
#pragma unroll 4
        for (int p = 0; p < NPOINT; ++p) {
            v2f ap, bp;
            ap.x = s_ey[p * PADW + offA];
            ap.y = 0.0f;
            bp.x = s_ex[p * PADW + offB];
            bp.y = 0.0f;
            v8f g = __builtin_amdgcn_wmma_f32_16x16x4_f32(
                false, ap, false, bp, (short)0, czero, false, false);
#pragma unroll
            for (int r = 0; r < 8; ++r) m[r] = fmaxf(m[r], g[r]);
        }

        // store: lanes 0-15 write rows ti*16+r, lanes 16-31 rows ti*16+8+r
#pragma unroll
        for (int r = 0; r < 8; ++r) {
            int row = ti * 16 + half * 8 + r;
            int col = tj * 16 + lj;
            ob[row * MASKN + col] = dbox[r];
            op[row * MASKN + col] = m[r];
        }
    }
}

extern "C" void kernel_launch(void* const* d_in, const int* in_sizes, int n_in,
                              void* d_out, int out_size, void* d_ws, size_t ws_size,
                              hipStream_t stream) {
    const float* boxes = (const float*)d_in[0];   // (B,4) f32
    const float* pts   = (const float*)d_in[1];   // (B,32,2) f32
    float* out = (float*)d_out;
    int B = in_sizes[0] / 4;                      // 256
    Zim_sam_mask_kernel<<<B, 128, 0, stream>>>(boxes, pts, out, B);
}